// SoftNCutsLoss_15487652069588
// MI455X (gfx1250) — compile-verified
//
#include <hip/hip_runtime.h>
#include <hip/hip_bf16.h>

// SoftNCutsLoss on MI455X (gfx1250, wave32).
//
// Wm is symmetric:  num_k = A_k^T Wm A_k,  den_k = A_k^T (Wm * 1).
// Both come from Y = Wm * B with B = [A0 A1 A2 A3 | ones | 0...] (P x 16),
// computed with V_WMMA_F32_16X16X4_F32 while Wm fragments are generated on
// the fly (one fused exp per entry). The kernel is VALU/trans bound
// (33.5M v_exp_f32); all operands stay f32 so there is no precision loss.
//
// This revision removes the divergent predicated B-loads (now unconditional
// load + v_cndmask), makes the inputs fetch wave-uniform (float4), and
// shares the (di,dj) spatial-distance terms between the lane's two K slots.

typedef __attribute__((ext_vector_type(2))) float v2f;
typedef __attribute__((ext_vector_type(8))) float v8f;

#define P_TOT   4096
#define N_BATCH 2
#define K_CH    4
#define QS      8                 // q-chunks per p-tile (parallelism: 4096 waves)
#define QLEN    (P_TOT / QS)      // 512 q per chunk

__global__ void __launch_bounds__(256)
softncuts_wmma_kernel(const float* __restrict__ labels,   // [2][4][4096]
                      const float* __restrict__ inputs,   // [2][1][4096]
                      float* __restrict__ Y) {            // [2][4096][5] accum
    const int lane = threadIdx.x & 31;
    const int wave = threadIdx.x >> 5;
    const int task = blockIdx.x * 8 + wave;          // 2 * 256 * QS tasks
    const int n     = task / (256 * QS);
    const int rem   = task % (256 * QS);
    const int ptile = rem / QS;
    const int qc    = rem % QS;

    const int p0 = ptile * 16;
    const int N  = lane & 15;        // column index of B/C fragment, row of A
    const int hi = lane >> 4;        // half-wave select
    const int ka = hi * 2;           // this lane's K pair: {0,1} or {2,3}

    const float* inp = inputs + n * P_TOT;
    // Always-valid label stream (lanes 4..15 re-read channel N&3; broadcast
    // from L0). Selected against bconst with a value select, no EXEC games.
    const float* labPtr = labels + (n * K_CH + (N & 3)) * P_TOT;
    const bool  isLab  = (N < 4);
    const float bconst = (N == 4) ? 1.0f : 0.0f;     // ones-column -> rowsum

    // A-matrix row owned by this lane (M = lane & 15 in both halves)
    const int   p    = p0 + N;
    const float in_p = inp[p];
    const int   pi = p >> 8, pj = (p >> 4) & 15, pk = p & 15;

    v8f c = {0.f, 0.f, 0.f, 0.f, 0.f, 0.f, 0.f, 0.f};

    const int qbase = qc * QLEN;
    #pragma unroll 4
    for (int it = 0; it < QLEN / 4; ++it) {
        const int q0 = qbase + it * 4;
        const int qa = q0 + ka;                       // qa even; qb = qa+1

        // Wave-uniform fetch of inputs[q0..q0+3] (single b128 load).
        const float4 qv = *(const float4*)(inp + q0);
        const float in_qa = hi ? qv.z : qv.x;
        const float in_qb = hi ? qv.w : qv.y;

        // Unconditional per-lane B loads, then value select.
        const float lab0 = labPtr[qa];
        const float lab1 = labPtr[qa + 1];

        // Spatial terms: qa/qb share (qi,qj); qk differs by exactly 1.
        const int qi = qa >> 8, qj = (qa >> 4) & 15, qk = qa & 15;
        const int di = pi - qi, dj = pj - qj;
        const int s2  = di * di + dj * dj;
        const int dka = pk - qk;
        const int dkb = dka - 1;
        const int sqa = s2 + dka * dka;
        const int sqb = s2 + dkb * dkb;

        const float da = in_p - in_qa;
        const float db = in_p - in_qb;
        const float arga = fmaf(da * da, -0.1f, (float)sqa * -0.25f);
        const float argb = fmaf(db * db, -0.1f, (float)sqb * -0.25f);
        const float ea = __expf(arga);
        const float eb = __expf(argb);

        v2f a, b;
        a.x = (sqa <= 25) ? ea : 0.0f;    // Wm[p, qa]
        a.y = (sqb <= 25) ? eb : 0.0f;    // Wm[p, qb]
        b.x = isLab ? lab0 : bconst;      // B[ka][N]
        b.y = isLab ? lab1 : bconst;      // B[ka+1][N]

        // D = A x B + C  ->  v_wmma_f32_16x16x4_f32
        c = __builtin_amdgcn_wmma_f32_16x16x4_f32(false, a, false, b,
                                                  (short)0, c, false, false);
    }

    // C/D layout: lane L, VGPR r -> row p0 + r + 8*hi, col N.
    // Only columns 0..4 carry signal (A0..A3 and the ones-column / rowsum).
    if (N < 5) {
        #pragma unroll
        for (int r = 0; r < 8; ++r) {
            int row = p0 + r + 8 * hi;
            atomicAdd(&Y[(n * P_TOT + row) * 5 + N], c[r]);
        }
    }
}

__global__ void __launch_bounds__(256)
softncuts_reduce_kernel(const float* __restrict__ labels,
                        const float* __restrict__ Y,
                        float* __restrict__ out) {
    __shared__ float red[256];
    __shared__ float accnum[N_BATCH][K_CH];
    __shared__ float accden[N_BATCH][K_CH];
    const int t = threadIdx.x;

    for (int n = 0; n < N_BATCH; ++n) {
        for (int k = 0; k < K_CH; ++k) {
            float s_num = 0.f, s_den = 0.f;
            for (int p = t; p < P_TOT; p += 256) {
                float a = labels[(n * K_CH + k) * P_TOT + p];
                s_num += a * Y[(n * P_TOT + p) * 5 + k];
                s_den += a * Y[(n * P_TOT + p) * 5 + 4];
            }
            red[t] = s_num; __syncthreads();
            for (int s = 128; s > 0; s >>= 1) {
                if (t < s) red[t] += red[t + s];
                __syncthreads();
            }
            if (t == 0) accnum[n][k] = red[0];
            __syncthreads();
            red[t] = s_den; __syncthreads();
            for (int s = 128; s > 0; s >>= 1) {
                if (t < s) red[t] += red[t + s];
                __syncthreads();
            }
            if (t == 0) accden[n][k] = red[0];
            __syncthreads();
        }
    }

    if (t < N_BATCH) {
        float loss = 0.f;
        for (int k = 0; k < K_CH; ++k)
            loss += accnum[t][k] / (accden[t][k] + 1e-8f);
        out[t] = (float)K_CH - loss;
    }
}

extern "C" void kernel_launch(void* const* d_in, const int* in_sizes, int n_in,
                              void* d_out, int out_size, void* d_ws, size_t ws_size,
                              hipStream_t stream) {
    (void)in_sizes; (void)n_in; (void)out_size; (void)ws_size;
    const float* labels = (const float*)d_in[0];   // (2,4,16,16,16) f32
    const float* inputs = (const float*)d_in[1];   // (2,1,16,16,16) f32
    float* out = (float*)d_out;                    // (2,) f32
    float* Y   = (float*)d_ws;                     // [2][4096][5] f32 accum

    const size_t ybytes = (size_t)N_BATCH * P_TOT * 5 * sizeof(float);
    hipMemsetAsync(d_ws, 0, ybytes, stream);       // graph-capture safe

    const int tasks  = N_BATCH * 256 * QS;         // 4096 waves
    const int blocks = tasks / 8;                  // 8 waves per block
    softncuts_wmma_kernel<<<dim3(blocks), dim3(256), 0, stream>>>(labels, inputs, Y);
    softncuts_reduce_kernel<<<dim3(1), dim3(256), 0, stream>>>(labels, Y, out);
}